// DynamicGC_41497974014274
// MI455X (gfx1250) — compile-verified
//
#include <hip/hip_runtime.h>

#define NNODES 50000
#define NEDGES 1600000
#define HD     256
#define LN_EPS 1e-5f

typedef float v2f __attribute__((ext_vector_type(2)));
typedef float v8f __attribute__((ext_vector_type(8)));
typedef unsigned int v4u __attribute__((ext_vector_type(4)));
typedef int v4i __attribute__((ext_vector_type(4)));
typedef int v8i __attribute__((ext_vector_type(8)));

// ---------------------------------------------------------------- utilities
__global__ void zero_kernel(float* __restrict__ p, int n) {
    int i = blockIdx.x * blockDim.x + threadIdx.x;
    if (i < n) p[i] = 0.0f;
}

__global__ void deg_kernel(const int* __restrict__ src, const int* __restrict__ dst,
                           float* __restrict__ degO, float* __restrict__ degI, int e) {
    int i = blockIdx.x * blockDim.x + threadIdx.x;
    if (i < e) {
        atomicAdd(degO + src[i], 1.0f);
        atomicAdd(degI + dst[i], 1.0f);
    }
}

__global__ void rsq_kernel(float* __restrict__ d, int n) {
    int i = blockIdx.x * blockDim.x + threadIdx.x;
    if (i < n) d[i] = rsqrtf(fmaxf(d[i], 1.0f));
}

// out[row,c] = x[row,c] * rsqO[row]
__global__ void scale_kernel(const float* __restrict__ x, const float* __restrict__ rsqO,
                             float* __restrict__ out, int n) {
    int i = blockIdx.x * blockDim.x + threadIdx.x;
    if (i < n) out[i] = x[i] * rsqO[i >> 8];   // HD == 256
}

// ------------------------------------------------------------- WMMA GEMM
// C[N x 256] = A[N x 256] @ W[256 x 256], fp32 via V_WMMA_F32_16X16X4_F32.
// Block = 8 waves; block owns a 64-col group; wave w owns row tile
// rowBlock*8+w (16 rows x 64 cols strip, 4 accumulator tiles).
// W panel (128 K-rows x 64 cols, 32 KB) is staged into LDS by the Tensor
// Data Mover (tensor_load_to_lds, TENSORcnt), issued by wave 0; B tiles are
// then read from LDS (ds_load), eliminating redundant global W traffic.
// A-layout (16x4 f32): lane[0:15]=M, VGPR0/1 = K{0,1}; lanes[16:31] = K{2,3}.
// C/D layout: VGPR r -> M=r (lanes 0-15), M=r+8 (lanes 16-31), N=lane&15.
__global__ __launch_bounds__(256) void gemm_wmma(const float* __restrict__ A,
                                                 const float* __restrict__ W,
                                                 float* __restrict__ C) {
    __shared__ float panel[128 * 64];            // 32 KB K-panel of W

    int colGroup = blockIdx.x & 3;               // 0..3  (4 * 64 cols = 256)
    int rowBlock = blockIdx.x >> 2;              // 0..390
    int wid  = threadIdx.x >> 5;                 // wave in block, 0..7
    int lane = threadIdx.x & 31;
    int rowTile = rowBlock * 8 + wid;            // 0..3127 (3125 valid)
    bool rowValid = (rowTile < NNODES / 16);     // wave-uniform

    int laneM = lane & 15;
    int hi    = lane >> 4;                       // 0 or 1

    v8f c0 = {}, c1 = {}, c2 = {}, c3 = {};

    for (int kk = 0; kk < HD; kk += 128) {
        __syncthreads();                         // prev-stage LDS reads done
        if (wid == 0) {
            // ---- Tensor DMA descriptor (CDNA5 ISA ch.8): 2D tile,
            // tile_dim0=64 (cols), tile_dim1=128 (K rows), 4-byte elems,
            // tensor_dim0_stride=256 floats (row pitch of W).
            unsigned long long gaddr =
                (unsigned long long)(const void*)(W + (size_t)kk * HD + colGroup * 64);
            unsigned lds_base = (unsigned)(unsigned long long)(const void*)panel;
            v4u g0 = { 1u,                                   // count=1, no gather
                       lds_base,                             // lds_addr
                       (unsigned)(gaddr & 0xffffffffu),      // global_addr[31:0]
                       (unsigned)(((gaddr >> 32) & 0x01ffffffu) | 0x80000000u) }; // [56:32] | type=2
            v8i g1 = { 0x00020000,              // data_size=2 (4 bytes), mask=0
                       (int)(256u << 16),       // tensor_dim0 = 256  (bits 79:48)
                       (int)(128u << 16),       // tensor_dim1 = 128  (bits 111:80)
                       (int)(64u  << 16),       // tile_dim0 = 64     (bits 127:112)
                       128,                     // tile_dim1 = 128    (bits 143:128)
                       256,                     // tensor_dim0_stride (bits 207:160)
                       0, 0 };
            v4i gz = { 0, 0, 0, 0 };
#if defined(__clang_major__) && (__clang_major__ >= 23)
            v8i gz8 = { 0, 0, 0, 0, 0, 0, 0, 0 };
            __builtin_amdgcn_tensor_load_to_lds(g0, g1, gz, gz, gz8, 0);
#else
            __builtin_amdgcn_tensor_load_to_lds(g0, g1, gz, gz, 0);
#endif
            __builtin_amdgcn_s_wait_tensorcnt((short)0);
        }
        __syncthreads();                         // panel visible to all waves

        if (rowValid) {                          // wave-uniform: EXEC all-ones
            const float* arow = A + (size_t)(rowTile * 16 + laneM) * HD + kk;
            #pragma unroll 4
            for (int k0 = 0; k0 < 128; k0 += 4) {
                int ka = k0 + 2 * hi;
                v2f a = *(const v2f*)(arow + ka);          // K = ka, ka+1
                const float* bp = panel + ka * 64 + laneM; // LDS B tiles
                v2f b0 = { bp[0],  bp[64]      };
                v2f b1 = { bp[16], bp[64 + 16] };
                v2f b2 = { bp[32], bp[64 + 32] };
                v2f b3 = { bp[48], bp[64 + 48] };
                c0 = __builtin_amdgcn_wmma_f32_16x16x4_f32(false, a, false, b0, (short)0, c0, false, false);
                c1 = __builtin_amdgcn_wmma_f32_16x16x4_f32(false, a, false, b1, (short)0, c1, false, false);
                c2 = __builtin_amdgcn_wmma_f32_16x16x4_f32(false, a, false, b2, (short)0, c2, false, false);
                c3 = __builtin_amdgcn_wmma_f32_16x16x4_f32(false, a, false, b3, (short)0, c3, false, false);
            }
        }
    }

    if (rowValid) {
        float* crow = C + (size_t)(rowTile * 16 + 8 * hi) * HD + colGroup * 64 + laneM;
        #pragma unroll
        for (int r = 0; r < 8; ++r) {
            crow[(size_t)r * HD + 0]  = c0[r];
            crow[(size_t)r * HD + 16] = c1[r];
            crow[(size_t)r * HD + 32] = c2[r];
            crow[(size_t)r * HD + 48] = c3[r];
        }
    }
}

// ------------------------------------------------------- edge aggregation
// One wave per edge: 32 lanes x (2 x float4) cover the 256-wide row.
// h and m (51 MB each) both fit in the 192 MB L2, so gathers + f32 atomics
// run at L2 speed; this phase is the end-to-end bottleneck.
__global__ __launch_bounds__(256) void scatter_kernel(const float* __restrict__ h,
                                                      const int* __restrict__ src,
                                                      const int* __restrict__ dst,
                                                      float* __restrict__ m) {
    int wave = (blockIdx.x * blockDim.x + threadIdx.x) >> 5;
    int lane = threadIdx.x & 31;
    if (wave >= NEDGES) return;
    int s = src[wave], d = dst[wave];
    const float4* hp = (const float4*)(h + (size_t)s * HD);
    float*        mp = m + (size_t)d * HD;
    float4 v0 = hp[lane];
    float4 v1 = hp[lane + 32];
    int o0 = lane * 4, o1 = (lane + 32) * 4;
    atomicAdd(mp + o0 + 0, v0.x); atomicAdd(mp + o0 + 1, v0.y);
    atomicAdd(mp + o0 + 2, v0.z); atomicAdd(mp + o0 + 3, v0.w);
    atomicAdd(mp + o1 + 0, v1.x); atomicAdd(mp + o1 + 1, v1.y);
    atomicAdd(mp + o1 + 2, v1.z); atomicAdd(mp + o1 + 3, v1.w);
}

// out[row,c] = relu(m[row,c]*rsqI[row] + b[c]) * rsqO[row]   (folds layer-2 in-scale)
__global__ void bias_relu_scale(const float* __restrict__ m, const float* __restrict__ rsqI,
                                const float* __restrict__ rsqO, const float* __restrict__ b,
                                float* __restrict__ out, int n) {
    int i = blockIdx.x * blockDim.x + threadIdx.x;
    if (i >= n) return;
    int row = i >> 8, c = i & 255;
    float v = fmaxf(m[i] * rsqI[row] + b[c], 0.0f);
    out[i] = v * rsqO[row];
}

// ------------------------------------------------------------- layer norm
// One wave per row; wave32 shuffle reductions over 256 features (8 per lane).
__global__ __launch_bounds__(256) void ln_kernel(const float* __restrict__ m,
                                                 const float* __restrict__ rsqI,
                                                 const float* __restrict__ b,
                                                 const float* __restrict__ gamma,
                                                 const float* __restrict__ beta,
                                                 float* __restrict__ out) {
    int wave = (blockIdx.x * blockDim.x + threadIdx.x) >> 5;
    int lane = threadIdx.x & 31;
    if (wave >= NNODES) return;
    const float* mr = m + (size_t)wave * HD + lane * 8;
    float ri = rsqI[wave];
    float4 p0 = *(const float4*)(mr);
    float4 p1 = *(const float4*)(mr + 4);
    float v[8];
    int cb = lane * 8;
    v[0] = p0.x * ri + b[cb + 0]; v[1] = p0.y * ri + b[cb + 1];
    v[2] = p0.z * ri + b[cb + 2]; v[3] = p0.w * ri + b[cb + 3];
    v[4] = p1.x * ri + b[cb + 4]; v[5] = p1.y * ri + b[cb + 5];
    v[6] = p1.z * ri + b[cb + 6]; v[7] = p1.w * ri + b[cb + 7];

    float s = 0.0f;
    #pragma unroll
    for (int j = 0; j < 8; ++j) s += v[j];
    #pragma unroll
    for (int off = 16; off > 0; off >>= 1) s += __shfl_xor(s, off, 32);
    float mu = s * (1.0f / HD);

    float s2 = 0.0f;
    #pragma unroll
    for (int j = 0; j < 8; ++j) { float t = v[j] - mu; s2 += t * t; }
    #pragma unroll
    for (int off = 16; off > 0; off >>= 1) s2 += __shfl_xor(s2, off, 32);
    float inv = rsqrtf(s2 * (1.0f / HD) + LN_EPS);

    float* orow = out + (size_t)wave * HD + cb;
    #pragma unroll
    for (int j = 0; j < 8; ++j)
        orow[j] = (v[j] - mu) * inv * gamma[cb + j] + beta[cb + j];
}

// ---------------------------------------------------------------- driver
extern "C" void kernel_launch(void* const* d_in, const int* in_sizes, int n_in,
                              void* d_out, int out_size, void* d_ws, size_t ws_size,
                              hipStream_t stream) {
    (void)in_sizes; (void)n_in; (void)out_size; (void)ws_size;
    const float* x     = (const float*)d_in[0];
    const float* W1    = (const float*)d_in[1];
    const float* b1    = (const float*)d_in[2];
    const float* W2    = (const float*)d_in[3];
    const float* b2    = (const float*)d_in[4];
    const float* gamma = (const float*)d_in[5];
    const float* beta  = (const float*)d_in[6];
    const int*   src   = (const int*)d_in[7];
    const int*   dst   = (const int*)d_in[8];
    float*       out   = (float*)d_out;

    float* bufA = (float*)d_ws;                       // N*H
    float* bufB = bufA + (size_t)NNODES * HD;         // N*H
    float* degO = bufB + (size_t)NNODES * HD;         // N  (becomes rsqrt(deg_out))
    float* degI = degO + NNODES;                      // N  (becomes rsqrt(deg_in))

    const int NH = NNODES * HD;                       // 12.8M, fits int
    const int THR = 256;
    const int rowBlocks     = (NNODES / 16 + 7) / 8;       // 391
    const int gemmBlocks    = rowBlocks * 4;               // 1564
    const int scatterBlocks = NEDGES / 8;                  // 8 edges (waves)/block

    // degrees -> rsqrt
    zero_kernel<<<(2 * NNODES + THR - 1) / THR, THR, 0, stream>>>(degO, 2 * NNODES);
    deg_kernel<<<(NEDGES + THR - 1) / THR, THR, 0, stream>>>(src, dst, degO, degI, NEDGES);
    rsq_kernel<<<(2 * NNODES + THR - 1) / THR, THR, 0, stream>>>(degO, 2 * NNODES);

    // layer 1
    scale_kernel<<<(NH + THR - 1) / THR, THR, 0, stream>>>(x, degO, bufA, NH);
    gemm_wmma<<<gemmBlocks, THR, 0, stream>>>(bufA, W1, bufB);
    zero_kernel<<<(NH + THR - 1) / THR, THR, 0, stream>>>(bufA, NH);
    scatter_kernel<<<scatterBlocks, THR, 0, stream>>>(bufB, src, dst, bufA);
    bias_relu_scale<<<(NH + THR - 1) / THR, THR, 0, stream>>>(bufA, degI, degO, b1, bufB, NH);

    // layer 2
    gemm_wmma<<<gemmBlocks, THR, 0, stream>>>(bufB, W2, bufA);
    zero_kernel<<<(NH + THR - 1) / THR, THR, 0, stream>>>(bufB, NH);
    scatter_kernel<<<scatterBlocks, THR, 0, stream>>>(bufA, src, dst, bufB);
    ln_kernel<<<NNODES / 8, THR, 0, stream>>>(bufB, degI, b2, gamma, beta, out);
}